// MPNN_block_14602888806939
// MI455X (gfx1250) — compile-verified
//
#include <hip/hip_runtime.h>
#include <hip/hip_bf16.h>
#include <stdint.h>

// ---------------------------------------------------------------------------
// GIN block: h = (1+eps)*x + scatter_add(relu(x[src]+edge_attr), dst)
//            out = h @ W + b
// N=100000, D=64, E=1600000. Memory-bound (~1.3 GB traffic -> ~56us @23.3TB/s).
// Edge phase = ~95% of traffic -> force native non-returning
// global_atomic_add_f32 (no CAS loop) via inline asm.
// GEMM is tiny (0.82 GFLOP) -> keep fp32, V_WMMA_F32_16X16X4_F32.
// ---------------------------------------------------------------------------

typedef __attribute__((ext_vector_type(2))) float v2f;
typedef __attribute__((ext_vector_type(8))) float v8f;

#define D 64

// Kernel 1: agg[i] = x[i] * (1 + eps)   (float4 vectorized)
__global__ void gin_init_self(const float* __restrict__ x,
                              const float* __restrict__ eps,
                              float* __restrict__ agg,
                              long long total4) {
  long long gid = (long long)blockIdx.x * blockDim.x + threadIdx.x;
  if (gid >= total4) return;
  float s = 1.0f + eps[0];
  const float4 xv = ((const float4*)x)[gid];
  float4 r;
  r.x = xv.x * s; r.y = xv.y * s; r.z = xv.z * s; r.w = xv.w * s;
  ((float4*)agg)[gid] = r;
}

// Kernel 2: per (edge, 4-feature chunk): atomic scatter-add relu(x[src]+ea).
// Four fire-and-forget hardware fp32 atomics share one address VGPR pair via
// offset immediates (VGLOBAL GV mode). Tracked on STOREcnt; S_ENDPGM's
// implicit wait-idle guarantees completion before the kernel retires.
__global__ void gin_edge_scatter(const float* __restrict__ x,
                                 const int64_t* __restrict__ ei,  // [2,E]
                                 const float* __restrict__ ea,
                                 float* __restrict__ agg,
                                 int E) {
  long long gid = (long long)blockIdx.x * blockDim.x + threadIdx.x;
  long long total = (long long)E * (D / 4);
  if (gid >= total) return;
  int e  = (int)(gid >> 4);          // D/4 == 16 chunks per edge
  int f  = (int)(gid & 15) * 4;
  long long dst = ei[e];
  long long src = ei[(long long)E + e];
  const float4 xv = *(const float4*)(x  + src * (long long)D + f);
  const float4 av = *(const float4*)(ea + (long long)e * D + f);
  float4 m;
  m.x = fmaxf(xv.x + av.x, 0.0f);
  m.y = fmaxf(xv.y + av.y, 0.0f);
  m.z = fmaxf(xv.z + av.z, 0.0f);
  m.w = fmaxf(xv.w + av.w, 0.0f);
  float* p = agg + dst * (long long)D + f;
  asm volatile(
      "global_atomic_add_f32 %0, %1, off\n\t"
      "global_atomic_add_f32 %0, %2, off offset:4\n\t"
      "global_atomic_add_f32 %0, %3, off offset:8\n\t"
      "global_atomic_add_f32 %0, %4, off offset:12"
      :
      : "v"(p), "v"(m.x), "v"(m.y), "v"(m.z), "v"(m.w)
      : "memory");
}

// Kernel 3: out = h @ W + b using V_WMMA_F32_16X16X4_F32.
// Block = 128 threads (4 waves). Block -> one 16-row M tile; wave w -> 16-col
// N tile w. K=64 consumed in 16 steps of K=4.
//
// fp32 16x16x4 fragment layouts (ISA 7.12.2):
//   A (16x4):  lanes 0-15 hold M=lane, v[0]=K0,v[1]=K1; lanes 16-31: K2,K3
//   B (4x16):  mirrored: v[0]: lanes0-15 row K0 / lanes16-31 row K2; v[1]: K1/K3
//   C/D (16x16): v[r]: lanes0-15 -> (M=r,     N=lane)
//                      lanes16-31 -> (M=r+8, N=lane-16)
__global__ void gin_gemm_wmma(const float* __restrict__ h,
                              const float* __restrict__ W,
                              const float* __restrict__ bias,
                              float* __restrict__ out) {
  const int lane  = threadIdx.x & 31;
  const int wave  = threadIdx.x >> 5;       // N tile 0..3
  const int mtile = blockIdx.x;
  const int mrow  = (lane & 15);            // row within tile for A
  const int khalf = (lane >> 4) * 2;        // 0 or 2
  const int ncol  = wave * 16 + (lane & 15);

  const float* __restrict__ hrow = h + ((long long)(mtile * 16 + mrow)) * D;

  v8f c = {};
#pragma unroll
  for (int k = 0; k < D; k += 4) {
    v2f a, bf;
    a.x  = hrow[k + khalf];
    a.y  = hrow[k + khalf + 1];
    bf.x = W[(k + khalf) * D + ncol];
    bf.y = W[(k + khalf + 1) * D + ncol];
    c = __builtin_amdgcn_wmma_f32_16x16x4_f32(
        /*neg_a=*/false, a, /*neg_b=*/false, bf,
        /*c_mod=*/(short)0, c, /*reuse_a=*/false, /*reuse_b=*/false);
  }

  const float bv = bias[ncol];
  const int mbase = mtile * 16 + ((lane >> 4) * 8);
#pragma unroll
  for (int r = 0; r < 8; ++r) {
    out[(long long)(mbase + r) * D + ncol] = c[r] + bv;
  }
}

extern "C" void kernel_launch(void* const* d_in, const int* in_sizes, int n_in,
                              void* d_out, int out_size, void* d_ws, size_t ws_size,
                              hipStream_t stream) {
  const float*   x    = (const float*)d_in[0];
  const int64_t* ei   = (const int64_t*)d_in[1];
  const float*   ea   = (const float*)d_in[2];
  const float*   eps  = (const float*)d_in[3];
  const float*   W    = (const float*)d_in[4];
  const float*   bias = (const float*)d_in[5];
  float*         out  = (float*)d_out;
  float*         agg  = (float*)d_ws;       // h accumulator: N*D floats (25.6 MB)

  const int N = in_sizes[0] / D;            // 100000
  const int E = in_sizes[1] / 2;            // 1600000

  // 1) seed accumulator with self term
  {
    long long total4 = (long long)N * D / 4;
    int blocks = (int)((total4 + 255) / 256);
    gin_init_self<<<blocks, 256, 0, stream>>>(x, eps, agg, total4);
  }
  // 2) edge scatter (hardware fp32 atomics)
  {
    long long total = (long long)E * (D / 4);
    int blocks = (int)((total + 255) / 256);
    gin_edge_scatter<<<blocks, 256, 0, stream>>>(x, ei, ea, agg, E);
  }
  // 3) WMMA projection  (N % 16 == 0)
  {
    int mtiles = N / 16;                    // 6250
    gin_gemm_wmma<<<mtiles, 128, 0, stream>>>(agg, W, bias, out);
  }
}